// Generator3DLUT_identity_86122684220302
// MI455X (gfx1250) — compile-verified
//
#include <hip/hip_runtime.h>

// ---------------------------------------------------------------------------
// Trilinear 3D-LUT apply (B=8, H=1080, W=1920, DIM=33), MI455X / gfx1250.
// Bandwidth-bound: ~398 MB streamed -> ~17us floor @ 23.3 TB/s.
// Strategy:
//   * Repack LUT to float4 {r,g,b,0} entries in d_ws -> 1 b128 load per corner.
//   * Stage input pixel tiles (1024 px x 3 planes = 12KB) into LDS with the
//     Tensor Data Mover (tensor_load_to_lds), double-buffered, waited with
//     s_wait_tensorcnt. Frees the VMEM pipe for the 8 b128 LUT gathers/px.
//   * float4 (pixel-quad) compute, b128 stores.
// ---------------------------------------------------------------------------

#define LUT_DIM   33
#define LUT_DIM2  (LUT_DIM * LUT_DIM)          // 1089
#define NLUT      (LUT_DIM * LUT_DIM * LUT_DIM)// 35937
#define IMG_H     1080
#define IMG_W     1920
#define IMG_HW    (IMG_H * IMG_W)              // 2073600
#define NBATCH    8
#define TPB       256
#define PIX_PER_TILE   (TPB * 4)               // 1024
#define TILES_PER_IMG  ((IMG_HW / 4) / TPB)    // 2025 (exact)
#define NUM_TILES      (NBATCH * TILES_PER_IMG)// 16200

#if __has_builtin(__builtin_amdgcn_tensor_load_to_lds) && \
    __has_builtin(__builtin_amdgcn_s_wait_tensorcnt)
#define USE_TDM 1
#else
#define USE_TDM 0
#endif

typedef unsigned int       u32;
typedef unsigned long long u64;
typedef u32 u32x4 __attribute__((ext_vector_type(4)));
typedef int i32x4 __attribute__((ext_vector_type(4)));
typedef int i32x8 __attribute__((ext_vector_type(8)));

union F4 { float4 v; float f[4]; };

// --- repack LUT: channel-planar [3][33^3] -> interleaved float4 [33^3] -------
__global__ void lut_repack_kernel(const float* __restrict__ LUT,
                                  float4* __restrict__ lut4) {
    int i = blockIdx.x * blockDim.x + threadIdx.x;
    if (i < NLUT) {
        float4 e;
        e.x = LUT[i];
        e.y = LUT[i + NLUT];
        e.z = LUT[i + 2 * NLUT];
        e.w = 0.0f;
        lut4[i] = e;
    }
}

#if USE_TDM
// Stage one tile (3 rows of 1024 floats, row stride = IMG_HW elements) into
// LDS as 12KB contiguous via the Tensor Data Mover. 2D descriptor, data_size=4B.
__device__ __forceinline__ void tdm_stage_tile(const float* x, long long tile,
                                               float* lds_dst) {
    long long img = tile / TILES_PER_IMG;
    long long off = img * (3LL * IMG_HW) +
                    (long long)(tile % TILES_PER_IMG) * PIX_PER_TILE;
    u64 ga  = (u64)(const void*)(x + off);
    u32 lds = (u32)(u64)(void*)lds_dst;   // low 32 bits of generic = LDS offset

    // D# group 0: count=1 | lds_addr | global_addr[56:0] | type=2
    u32x4 g0 = { 1u,
                 lds,
                 (u32)(ga & 0xFFFFFFFFull),
                 (u32)((ga >> 32) & 0x1FFFFFFull) | (2u << 30) };

    // D# group 1 (bit-packed per ISA 8.4):
    //  w0: workgroup_mask=0, data_size=2 (4B)
    //  w1[31:16]=tensor_dim0[15:0]; w2[15:0]=tensor_dim0[31:16]; w2[31:16]=tensor_dim1=3
    //  w3[31:16]=tile_dim0=1024; w4[15:0]=tile_dim1=3
    //  w5=tensor_dim0_stride[31:0]=IMG_HW; rest 0
    i32x8 g1 = { (int)(2u << 16),
                 (int)(((u32)IMG_HW & 0xFFFFu) << 16),
                 (int)((((u32)IMG_HW >> 16) & 0xFFFFu) | (3u << 16)),
                 (int)((u32)PIX_PER_TILE << 16),
                 3,
                 (int)IMG_HW,
                 0,
                 0 };
    i32x4 z4 = { 0, 0, 0, 0 };
    i32x8 z8 = { 0, 0, 0, 0, 0, 0, 0, 0 };
    // 6-arg form (clang-23 / therock-10.0 toolchain).
    __builtin_amdgcn_tensor_load_to_lds(g0, g1, z4, z4, z8, 0);
}
#endif

template <bool PACKED>
__device__ __forceinline__ float4 lut_fetch(const float4* __restrict__ lut4,
                                            const float* __restrict__ LUT,
                                            int e) {
    if (PACKED) {
        return lut4[e];                       // single global_load_b128
    } else {
        float4 c;
        c.x = LUT[e];
        c.y = LUT[e + NLUT];
        c.z = LUT[e + 2 * NLUT];
        c.w = 0.0f;
        return c;
    }
}

template <bool PACKED>
__global__ __launch_bounds__(TPB) void lut_apply_kernel(
    const float*  __restrict__ x,
    const float4* __restrict__ lut4,
    const float*  __restrict__ LUT,
    float*        __restrict__ out) {
    const float inv_bin = 32.0f / 1.000001f;  // 1/binsize
    const int t = threadIdx.x;

#if USE_TDM
    __shared__ float sbuf[2][3 * PIX_PER_TILE];   // 24 KB double buffer
    // Scalar-branch wave selection: TDM ignores EXEC, must branch around it.
    const bool wave0 = (__builtin_amdgcn_readfirstlane((int)threadIdx.x) == 0);
    int cur = 0;
    {
        long long first = blockIdx.x;
        if (first < NUM_TILES && wave0) tdm_stage_tile(x, first, &sbuf[0][0]);
    }
#endif

    for (long long tile = blockIdx.x; tile < NUM_TILES; tile += gridDim.x) {
        const long long nxt = tile + gridDim.x;
        const bool hasNext  = (nxt < NUM_TILES);
        F4 R, G, Bb;

#if USE_TDM
        __syncthreads();   // all readers of buf[cur^1] (prev iter) are done
        if (wave0) {
            if (hasNext) {
                tdm_stage_tile(x, nxt, &sbuf[cur ^ 1][0]);
                __builtin_amdgcn_s_wait_tensorcnt(1);   // tile 'tile' landed
            } else {
                __builtin_amdgcn_s_wait_tensorcnt(0);
            }
        }
        __syncthreads();   // tile data visible to all waves
        R.v  = *(const float4*)&sbuf[cur][0 * PIX_PER_TILE + t * 4];
        G.v  = *(const float4*)&sbuf[cur][1 * PIX_PER_TILE + t * 4];
        Bb.v = *(const float4*)&sbuf[cur][2 * PIX_PER_TILE + t * 4];
        cur ^= 1;
#endif

        long long img  = tile / TILES_PER_IMG;
        long long base = img * (3LL * IMG_HW) +
                         (long long)(tile % TILES_PER_IMG) * PIX_PER_TILE +
                         t * 4;
#if !USE_TDM
        R.v  = *(const float4*)(x + base);
        G.v  = *(const float4*)(x + base + IMG_HW);
        Bb.v = *(const float4*)(x + base + 2 * IMG_HW);
#endif

        F4 oR, oG, oB;
#pragma unroll
        for (int j = 0; j < 4; ++j) {
            float rf = R.f[j]  * inv_bin;
            float gf = G.f[j]  * inv_bin;
            float bf = Bb.f[j] * inv_bin;
            int rid = (int)floorf(rf);
            int gid = (int)floorf(gf);
            int bid = (int)floorf(bf);
            rid = rid < 0 ? 0 : (rid > LUT_DIM - 2 ? LUT_DIM - 2 : rid);
            gid = gid < 0 ? 0 : (gid > LUT_DIM - 2 ? LUT_DIM - 2 : gid);
            bid = bid < 0 ? 0 : (bid > LUT_DIM - 2 ? LUT_DIM - 2 : bid);
            float rd = rf - (float)rid;
            float gd = gf - (float)gid;
            float bd = bf - (float)bid;

            int e = (bid * LUT_DIM + gid) * LUT_DIM + rid;
            float4 c000 = lut_fetch<PACKED>(lut4, LUT, e);
            float4 c001 = lut_fetch<PACKED>(lut4, LUT, e + 1);
            float4 c010 = lut_fetch<PACKED>(lut4, LUT, e + LUT_DIM);
            float4 c011 = lut_fetch<PACKED>(lut4, LUT, e + LUT_DIM + 1);
            float4 c100 = lut_fetch<PACKED>(lut4, LUT, e + LUT_DIM2);
            float4 c101 = lut_fetch<PACKED>(lut4, LUT, e + LUT_DIM2 + 1);
            float4 c110 = lut_fetch<PACKED>(lut4, LUT, e + LUT_DIM2 + LUT_DIM);
            float4 c111 = lut_fetch<PACKED>(lut4, LUT, e + LUT_DIM2 + LUT_DIM + 1);

            float wr1 = rd, wr0 = 1.0f - rd;
            float wg1 = gd, wg0 = 1.0f - gd;
            float wb1 = bd, wb0 = 1.0f - bd;
            float w00 = wb0 * wg0, w01 = wb0 * wg1;
            float w10 = wb1 * wg0, w11 = wb1 * wg1;
            float w000 = w00 * wr0, w001 = w00 * wr1;
            float w010 = w01 * wr0, w011 = w01 * wr1;
            float w100 = w10 * wr0, w101 = w10 * wr1;
            float w110 = w11 * wr0, w111 = w11 * wr1;

            float r_ = w000 * c000.x;
            r_ = fmaf(w001, c001.x, r_); r_ = fmaf(w010, c010.x, r_);
            r_ = fmaf(w011, c011.x, r_); r_ = fmaf(w100, c100.x, r_);
            r_ = fmaf(w101, c101.x, r_); r_ = fmaf(w110, c110.x, r_);
            r_ = fmaf(w111, c111.x, r_);

            float g_ = w000 * c000.y;
            g_ = fmaf(w001, c001.y, g_); g_ = fmaf(w010, c010.y, g_);
            g_ = fmaf(w011, c011.y, g_); g_ = fmaf(w100, c100.y, g_);
            g_ = fmaf(w101, c101.y, g_); g_ = fmaf(w110, c110.y, g_);
            g_ = fmaf(w111, c111.y, g_);

            float b_ = w000 * c000.z;
            b_ = fmaf(w001, c001.z, b_); b_ = fmaf(w010, c010.z, b_);
            b_ = fmaf(w011, c011.z, b_); b_ = fmaf(w100, c100.z, b_);
            b_ = fmaf(w101, c101.z, b_); b_ = fmaf(w110, c110.z, b_);
            b_ = fmaf(w111, c111.z, b_);

            oR.f[j] = r_; oG.f[j] = g_; oB.f[j] = b_;
        }

        *(float4*)(out + base)              = oR.v;
        *(float4*)(out + base + IMG_HW)     = oG.v;
        *(float4*)(out + base + 2 * IMG_HW) = oB.v;
    }
}

extern "C" void kernel_launch(void* const* d_in, const int* in_sizes, int n_in,
                              void* d_out, int out_size, void* d_ws,
                              size_t ws_size, hipStream_t stream) {
    (void)in_sizes; (void)n_in; (void)out_size;
    const float* x   = (const float*)d_in[0];
    const float* LUT = (const float*)d_in[1];
    float* out = (float*)d_out;

    const int grid = 2048;  // 8 waves/block, grid-stride over 16200 tiles

    if (d_ws && ws_size >= (size_t)NLUT * sizeof(float4)) {
        float4* lut4 = (float4*)d_ws;
        lut_repack_kernel<<<(NLUT + TPB - 1) / TPB, TPB, 0, stream>>>(LUT, lut4);
        lut_apply_kernel<true><<<grid, TPB, 0, stream>>>(x, lut4, LUT, out);
    } else {
        lut_apply_kernel<false><<<grid, TPB, 0, stream>>>(x, nullptr, LUT, out);
    }
}